// RoIAlignRotatedWrapper_38895223832978
// MI455X (gfx1250) — compile-verified
//
#include <hip/hip_runtime.h>

// ---------------- problem constants (from reference) ----------------
#define N_      2
#define C_      256
#define H_      200
#define W_      200
#define HW_     40000          // H_*W_
#define R_      1000
#define OUT_    7
#define NBINS   49             // 7*7
#define SCALE_  0.25f
#define EPS_    1e-8f
#define DEG2RAD 0.017453292519943295f

typedef __attribute__((ext_vector_type(4))) unsigned int u32x4;
typedef __attribute__((ext_vector_type(4))) int          i32x4;
typedef __attribute__((ext_vector_type(8))) int          i32x8;

#if __has_builtin(__builtin_amdgcn_tensor_load_to_lds)
#define CDNA5_HAS_TDM 1
#else
#define CDNA5_HAS_TDM 0
#endif

static __device__ __forceinline__ void wait_tensorcnt0() {
#if __has_builtin(__builtin_amdgcn_s_wait_tensorcnt)
  __builtin_amdgcn_s_wait_tensorcnt(0);
#else
  asm volatile("s_wait_tensorcnt 0x0" ::: "memory");
#endif
}

// =====================================================================
// Kernel 1: NCHW -> NHWC transpose, 32x32 tiles via LDS.
// TDM loads the tile global->LDS (pad_enable gives stride-33 rows, i.e.
// bank-conflict-free for the transposed read), then all 8 waves write
// coalesced 128B lines to the NHWC workspace.
// grid = (HW_/32, (N_*C_)/32), block = 256
// =====================================================================
__global__ void nchw_to_nhwc_kernel(const float* __restrict__ feat,
                                    float* __restrict__ ft) {
  __shared__ float tile[32 * 33];   // row stride 33 dwords (TDM padding)

  const int p0  = blockIdx.x * 32;        // pixel tile base (flat y*W+x)
  const int nc0 = blockIdx.y * 32;        // (n*C + c) tile base
  const int t   = threadIdx.x;
  const int tx  = t & 31;
  const int ty  = t >> 5;

#if CDNA5_HAS_TDM
  if (t < 32) {                      // wave 0 issues the tensor DMA once
    const unsigned long long ga =
        (unsigned long long)(uintptr_t)(feat + (size_t)nc0 * HW_ + p0);
    const unsigned int lds = (unsigned int)(uintptr_t)&tile[0];

    u32x4 g0;
    g0[0] = 1u;                                     // count=1, load-descriptor
    g0[1] = lds;                                    // lds_addr (bytes)
    g0[2] = (unsigned int)(ga & 0xFFFFFFFFu);       // global_addr[31:0]
    g0[3] = (unsigned int)((ga >> 32) & 0x01FFFFFFu)// global_addr[56:32]
            | (2u << 30);                           // type = 2 ("image")

    i32x8 g1;
    // data_size=2 (4B) | pad_enable | pad_interval=4 (every 32 dwords),
    // pad_amount=0 (1 dword) -> LDS row stride 33 dwords
    g1[0] = (2 << 16) | (1 << 20) | (4 << 22);
    g1[1] = (int)(32u << 16);        // tensor_dim0 = 32 (bits 63:48)
    g1[2] = (int)(32u << 16);        // tensor_dim0 hi=0; tensor_dim1 = 32
    g1[3] = (int)(32u << 16);        // tensor_dim1 hi=0; tile_dim0 = 32
    g1[4] = 32;                      // tile_dim1 = 32, tile_dim2 = 0
    g1[5] = HW_;                     // tensor_dim0_stride = 40000 elements
    g1[6] = 0;                       // stride0 hi = 0; dim1_stride lo16 = 0
    g1[7] = 0;

    i32x4 gz4 = {0, 0, 0, 0};
    i32x8 gz8 = {0, 0, 0, 0, 0, 0, 0, 0};
    __builtin_amdgcn_tensor_load_to_lds(g0, g1, gz4, gz4, gz8, 0);
    wait_tensorcnt0();
  }
#else
  // fallback: cooperative coalesced load (reads along W are contiguous)
  #pragma unroll
  for (int j = 0; j < 4; ++j) {
    const int ch = ty + 8 * j;
    tile[ch * 33 + tx] = feat[(size_t)(nc0 + ch) * HW_ + p0 + tx];
  }
#endif
  __syncthreads();

  // write NHWC: ft[(n*HW + p)*C + c]; lanes sweep c -> coalesced
  const int n  = nc0 >> 8;           // C_ = 256
  const int c0 = nc0 & 255;
  #pragma unroll
  for (int j = 0; j < 4; ++j) {
    const int p = ty + 8 * j;
    ft[(size_t)(n * HW_ + p0 + p) * C_ + c0 + tx] = tile[tx * 33 + p];
  }
}

// =====================================================================
// Kernel 2: rotated RoIAlign. One block per roi, thread = channel.
// LAYOUT 0: gather from NHWC workspace (coalesced).  LAYOUT 1: NCHW direct.
// =====================================================================
template <int LAYOUT>
__global__ void roi_align_rot_kernel(const float* __restrict__ feat,
                                     const float* __restrict__ rois,
                                     float* __restrict__ out) {
  __shared__ float4 s_w[NBINS * 4];      // per (bin,subsample) weights *0.25
  __shared__ int4   s_o[NBINS * 4];      // per (bin,subsample) corner offsets
  __shared__ float  s_acc[NBINS * 257];  // [bin][c], padded stride 257

  const int r = blockIdx.x;
  const int t = threadIdx.x;             // = channel c

  // ---- uniform roi parameters (cheap, every thread) ----
  const float* roi = rois + r * 6;
  const int   b  = (int)roi[0];
  const float cw = roi[1] * SCALE_ - 0.5f;
  const float ch = roi[2] * SCALE_ - 0.5f;
  const float rw = fmaxf(roi[3] * SCALE_, 1.0f);
  const float rh = fmaxf(roi[4] * SCALE_, 1.0f);
  const float th = roi[5] * DEG2RAD;
  const float cs = __cosf(th) , sn = __sinf(th);
  const float bh = rh * (1.0f / OUT_);
  const float bw = rw * (1.0f / OUT_);

  // ---- phase 1: precompute 49*4 sample params into LDS ----
  if (t < NBINS * 4) {
    const int bin = t >> 2, s = t & 3;
    const int oh = bin / OUT_, ow = bin % OUT_;
    const int sy = s >> 1, sx = s & 1;

    const float yy = -0.5f * rh + ((float)oh + ((float)sy + 0.5f) * 0.5f) * bh;
    const float xx = -0.5f * rw + ((float)ow + ((float)sx + 0.5f) * 0.5f) * bw;
    float y = yy * cs - xx * sn + ch;
    float x = yy * sn + xx * cs + cw;

    const bool valid = (y >= -1.0f) && (y <= (float)H_) &&
                       (x >= -1.0f) && (x <= (float)W_);
    y = fmaxf(y, 0.0f);
    x = fmaxf(x, 0.0f);

    int yl0 = (int)floorf(y), xl0 = (int)floorf(x);
    int ylo, yhi, xlo, xhi;
    float yv = y, xv = x;
    if (yl0 >= H_ - 1) { ylo = yhi = H_ - 1; yv = (float)(H_ - 1); }
    else               { ylo = yl0; yhi = yl0 + 1; }
    if (xl0 >= W_ - 1) { xlo = xhi = W_ - 1; xv = (float)(W_ - 1); }
    else               { xlo = xl0; xhi = xl0 + 1; }

    const float ly = yv - (float)ylo, lx = xv - (float)xlo;
    const float hy = 1.0f - ly,       hx = 1.0f - lx;
    const float m  = valid ? 0.25f : 0.0f;   // fold /(SR*SR) + mask
    s_w[t] = make_float4(hy * hx * m, hy * lx * m, ly * hx * m, ly * lx * m);

    const int mul = (LAYOUT == 0) ? C_ : 1;  // NHWC: pixel*C; NCHW: pixel
    s_o[t] = make_int4((ylo * W_ + xlo) * mul, (ylo * W_ + xhi) * mul,
                       (yhi * W_ + xlo) * mul, (yhi * W_ + xhi) * mul);
  }
  __syncthreads();

  // ---- phase 2: accumulate over bins; lanes sweep channels ----
  const float* base = (LAYOUT == 0)
      ? feat + (size_t)b * HW_ * C_ + t           // NHWC: + pix*C
      : feat + (size_t)(b * C_ + t) * HW_;        // NCHW: + pix

  for (int bin = 0; bin < NBINS; ++bin) {
    float acc = 0.0f;
    #pragma unroll
    for (int s = 0; s < 4; ++s) {
      const float4 w = s_w[bin * 4 + s];
      const int4   o = s_o[bin * 4 + s];
      acc += w.x * base[o.x] + w.y * base[o.y] +
             w.z * base[o.z] + w.w * base[o.w];
    }
    if (bin + 1 < NBINS)                      // global_prefetch_b8
      __builtin_prefetch(&base[s_o[(bin + 1) * 4].x], 0, 3);
    if (!__builtin_isfinite(acc)) acc = EPS_; // nan_to_num
    s_acc[bin * 257 + t] = acc;
  }
  __syncthreads();

  // ---- phase 3: coalesced writeout in [c][bin] order ----
  float* orow = out + (size_t)r * (C_ * NBINS);
  #pragma unroll 7
  for (int k = 0; k < NBINS; ++k) {
    const int idx = k * 256 + t;              // 0 .. 12543
    const int c   = idx / NBINS;
    const int bin = idx - c * NBINS;
    orow[idx] = s_acc[bin * 257 + c];
  }
}

// =====================================================================
extern "C" void kernel_launch(void* const* d_in, const int* in_sizes, int n_in,
                              void* d_out, int out_size, void* d_ws, size_t ws_size,
                              hipStream_t stream) {
  const float* feat = (const float*)d_in[0];   // [2,256,200,200] f32
  const float* rois = (const float*)d_in[1];   // [1000,6] f32
  float*       out  = (float*)d_out;           // [1000,256,7,7] f32

  const size_t need = (size_t)N_ * C_ * HW_ * sizeof(float);  // 81.92 MB
  if (d_ws && ws_size >= need) {
    float* ft = (float*)d_ws;
    dim3 tg(HW_ / 32, (N_ * C_) / 32);         // 1250 x 16 tiles
    nchw_to_nhwc_kernel<<<tg, 256, 0, stream>>>(feat, ft);
    roi_align_rot_kernel<0><<<R_, 256, 0, stream>>>(ft, rois, out);
  } else {
    roi_align_rot_kernel<1><<<R_, 256, 0, stream>>>(feat, rois, out);
  }
}